// SpGraphAttentionLayer_6571299963295
// MI455X (gfx1250) — compile-verified
//
#include <hip/hip_runtime.h>

// Problem constants (from reference)
#define B_    4
#define N_    10000
#define E_    320000
#define INF_  256
#define ATTD_ 256
#define H_    8
#define DK_   32
#define ALPHA_ 0.2f

typedef __attribute__((ext_vector_type(2)))  float    v2f;
typedef __attribute__((ext_vector_type(8)))  float    v8f;
typedef __attribute__((ext_vector_type(16))) _Float16 v16h;

// ---------------------------------------------------------------------------
// Kernel 0: one-time W transpose (256x256, 256 KB) so WMMA B fragments become
// contiguous b64 loads:  Wt[n*256 + k] = W[k*256 + n].
// ---------------------------------------------------------------------------
__global__ __launch_bounds__(256) void gat_transpose_w(const float* __restrict__ W,
                                                       float* __restrict__ Wt) {
  const int idx = blockIdx.x * blockDim.x + threadIdx.x;
  if (idx >= ATTD_ * INF_) return;
  const int k = idx >> 8;
  const int n = idx & 255;
  Wt[(size_t)n * INF_ + k] = W[idx];
}

// ---------------------------------------------------------------------------
// Kernel 1: wx = x @ W  (M=40000, N=256, K=256), fp32 WMMA (16x16x4).
// Block = 256 threads = 8 waves, owns a 32-row strip of x staged in LDS.
// Wave w owns columns [w*32, w*32+32): a 2(M) x 2(N) grid of 16x16 tiles.
// Per K-step: 2 ds_load_b64 (A) + 2 global_load_b64 (Bt) -> 4 WMMAs.
// ---------------------------------------------------------------------------
__global__ __launch_bounds__(256) void gat_gemm_wx(const float* __restrict__ x,
                                                   const float* __restrict__ Wt,
                                                   float* __restrict__ wx) {
  __shared__ float lA[32 * INF_];   // 32 KB

  const int tid  = threadIdx.x;
  const int lane = tid & 31;
  const int wave = tid >> 5;
  const int m0   = blockIdx.x * 32;
  const int n0   = wave * 32;
  const int row  = lane & 15;   // M within tile (A) / N within tile (B,C,D)
  const int hi   = lane >> 4;   // lane half: K pair select (A/B), M+8 (C/D)
  const int koff = hi * 2;

  // Stage the 32x256 A strip: 8192 floats, 8 x float4 per thread, coalesced.
  {
    const float4* gx = (const float4*)(x + (size_t)m0 * INF_);
    float4*       lx = (float4*)lA;
#pragma unroll
    for (int i = 0; i < 8; ++i) {
      lx[tid + i * 256] = gx[tid + i * 256];
    }
  }
  __syncthreads();

  v8f acc00 = {}, acc01 = {}, acc10 = {}, acc11 = {};

  // Bt rows for this wave's two N-tiles (contiguous in K after transpose).
  const float* bt0 = Wt + (size_t)(n0 + row) * INF_ + koff;
  const float* bt1 = Wt + (size_t)(n0 + 16 + row) * INF_ + koff;
  const float* la0 = lA + (size_t)row * INF_ + koff;
  const float* la1 = lA + (size_t)(row + 16) * INF_ + koff;

  for (int k = 0; k < INF_; k += 4) {
    // A fragments (16x4 layout: lanes 0-15 K={0,1}, lanes 16-31 K={2,3})
    const float2 a0v = *(const float2*)(la0 + k);
    const float2 a1v = *(const float2*)(la1 + k);
    v2f a0; a0[0] = a0v.x; a0[1] = a0v.y;
    v2f a1; a1[0] = a1v.x; a1[1] = a1v.y;
    // B fragments (4x16, N striped over lane halves like A)
    const float2 b0v = *(const float2*)(bt0 + k);
    const float2 b1v = *(const float2*)(bt1 + k);
    v2f b0; b0[0] = b0v.x; b0[1] = b0v.y;
    v2f b1; b1[0] = b1v.x; b1[1] = b1v.y;

    acc00 = __builtin_amdgcn_wmma_f32_16x16x4_f32(false, a0, false, b0,
                                                  (short)0, acc00, false, false);
    acc01 = __builtin_amdgcn_wmma_f32_16x16x4_f32(false, a0, false, b1,
                                                  (short)0, acc01, false, false);
    acc10 = __builtin_amdgcn_wmma_f32_16x16x4_f32(false, a1, false, b0,
                                                  (short)0, acc10, false, false);
    acc11 = __builtin_amdgcn_wmma_f32_16x16x4_f32(false, a1, false, b1,
                                                  (short)0, acc11, false, false);
  }

  // C/D layout: VGPR r -> M=r (lanes 0-15) / M=r+8 (lanes 16-31), N=lane&15.
  const int orow = hi * 8;
#pragma unroll
  for (int r = 0; r < 8; ++r) {
    float* o0 = wx + (size_t)(m0 + orow + r) * ATTD_ + row;
    o0[n0]      = acc00[r];
    o0[n0 + 16] = acc01[r];
    float* o1 = wx + (size_t)(m0 + 16 + orow + r) * ATTD_ + row;
    o1[n0]      = acc10[r];
    o1[n0 + 16] = acc11[r];
  }
}

// ---------------------------------------------------------------------------
// Kernel 2: per-(node,head) scores  s_src = <wx_head, a[:32]>, s_dst = <., a[32:]>
// ---------------------------------------------------------------------------
__global__ __launch_bounds__(256) void gat_node_scores(const float* __restrict__ wx,
                                                       const float* __restrict__ a,
                                                       float* __restrict__ s_src,
                                                       float* __restrict__ s_dst) {
  const int idx = blockIdx.x * blockDim.x + threadIdx.x;  // (b*N+n)*H + h
  if (idx >= B_ * N_ * H_) return;
  const int h    = idx & (H_ - 1);
  const int node = idx >> 3;
  const float* p = wx + (size_t)node * ATTD_ + h * DK_;
  float ss = 0.f, sd = 0.f;
#pragma unroll
  for (int k = 0; k < DK_; ++k) {
    const float v = p[k];
    ss += v * a[k];
    sd += v * a[DK_ + k];
  }
  s_src[idx] = ss;
  s_dst[idx] = sd;
}

// ---------------------------------------------------------------------------
// Monotonic uint key for float atomic-max (total order incl. negatives).
// ---------------------------------------------------------------------------
__device__ __forceinline__ unsigned f2ord(float f) {
  const unsigned u = __float_as_uint(f);
  return (u & 0x80000000u) ? ~u : (u | 0x80000000u);
}
__device__ __forceinline__ float ord2f(unsigned k) {
  const unsigned u = (k & 0x80000000u) ? (k ^ 0x80000000u) : ~k;
  return __uint_as_float(u);
}

__global__ __launch_bounds__(256) void gat_init_seg(unsigned* __restrict__ segmax,
                                                    float* __restrict__ segsum) {
  const int idx = blockIdx.x * blockDim.x + threadIdx.x;
  if (idx >= B_ * N_ * H_) return;
  segmax[idx] = 0u;   // key-space minimum (<= key(-inf))
  segsum[idx] = 0.f;
}

// ---------------------------------------------------------------------------
// Kernel 3: per-edge leaky-relu score -> att[], segment max via atomic key-max
// ---------------------------------------------------------------------------
__global__ __launch_bounds__(256) void gat_edge_score(const int* __restrict__ edge,
                                                      const float* __restrict__ s_src,
                                                      const float* __restrict__ s_dst,
                                                      float* __restrict__ att,
                                                      unsigned* __restrict__ segmax) {
  const int idx = blockIdx.x * blockDim.x + threadIdx.x;  // b*E + e
  if (idx >= B_ * E_) return;
  const int b = idx / E_;
  const int e = idx - b * E_;
  const int src = edge[(size_t)b * 2 * E_ + e];
  const int dst = edge[(size_t)b * 2 * E_ + E_ + e];
  const float4* ps = (const float4*)(s_src + (size_t)(b * N_ + src) * H_);
  const float4* pd = (const float4*)(s_dst + (size_t)(b * N_ + dst) * H_);
  unsigned* pm     = segmax + (size_t)(b * N_ + src) * H_;
  float4* po       = (float4*)(att + (size_t)idx * H_);

  float v[H_];
  {
    const float4 s0 = ps[0], s1 = ps[1], d0 = pd[0], d1 = pd[1];
    v[0] = s0.x + d0.x; v[1] = s0.y + d0.y; v[2] = s0.z + d0.z; v[3] = s0.w + d0.w;
    v[4] = s1.x + d1.x; v[5] = s1.y + d1.y; v[6] = s1.z + d1.z; v[7] = s1.w + d1.w;
  }
#pragma unroll
  for (int h = 0; h < H_; ++h) v[h] = (v[h] >= 0.f) ? v[h] : ALPHA_ * v[h];
  po[0] = make_float4(v[0], v[1], v[2], v[3]);
  po[1] = make_float4(v[4], v[5], v[6], v[7]);
#pragma unroll
  for (int h = 0; h < H_; ++h) atomicMax(pm + h, f2ord(v[h]));
}

// ---------------------------------------------------------------------------
// Kernel 4: ex = exp(score - segmax[seg]); segment sum via atomicAdd
// ---------------------------------------------------------------------------
__global__ __launch_bounds__(256) void gat_edge_exp(const int* __restrict__ edge,
                                                    const unsigned* __restrict__ segmax,
                                                    float* __restrict__ att,
                                                    float* __restrict__ segsum) {
  const int idx = blockIdx.x * blockDim.x + threadIdx.x;
  if (idx >= B_ * E_) return;
  const int b = idx / E_;
  const int e = idx - b * E_;
  const int src = edge[(size_t)b * 2 * E_ + e];
  const uint4* pm = (const uint4*)(segmax + (size_t)(b * N_ + src) * H_);
  float* psum     = segsum + (size_t)(b * N_ + src) * H_;
  float4* po      = (float4*)(att + (size_t)idx * H_);

  const float4 p0 = po[0], p1 = po[1];
  const uint4  m0 = pm[0], m1 = pm[1];
  float ex[H_];
  ex[0] = __expf(p0.x - ord2f(m0.x));
  ex[1] = __expf(p0.y - ord2f(m0.y));
  ex[2] = __expf(p0.z - ord2f(m0.z));
  ex[3] = __expf(p0.w - ord2f(m0.w));
  ex[4] = __expf(p1.x - ord2f(m1.x));
  ex[5] = __expf(p1.y - ord2f(m1.y));
  ex[6] = __expf(p1.z - ord2f(m1.z));
  ex[7] = __expf(p1.w - ord2f(m1.w));
  po[0] = make_float4(ex[0], ex[1], ex[2], ex[3]);
  po[1] = make_float4(ex[4], ex[5], ex[6], ex[7]);
#pragma unroll
  for (int h = 0; h < H_; ++h) atomicAdd(psum + h, ex[h]);
}

// ---------------------------------------------------------------------------
// Kernel 5: attention = ex / (segsum[seg] + 1e-16)
// ---------------------------------------------------------------------------
__global__ __launch_bounds__(256) void gat_edge_norm(const int* __restrict__ edge,
                                                     const float* __restrict__ segsum,
                                                     float* __restrict__ att) {
  const int idx = blockIdx.x * blockDim.x + threadIdx.x;
  if (idx >= B_ * E_) return;
  const int b = idx / E_;
  const int e = idx - b * E_;
  const int src = edge[(size_t)b * 2 * E_ + e];
  const float4* pq = (const float4*)(segsum + (size_t)(b * N_ + src) * H_);
  float4* po       = (float4*)(att + (size_t)idx * H_);
  const float4 q0 = pq[0], q1 = pq[1];
  float4 p0 = po[0], p1 = po[1];
  p0.x /= (q0.x + 1e-16f); p0.y /= (q0.y + 1e-16f);
  p0.z /= (q0.z + 1e-16f); p0.w /= (q0.w + 1e-16f);
  p1.x /= (q1.x + 1e-16f); p1.y /= (q1.y + 1e-16f);
  p1.z /= (q1.z + 1e-16f); p1.w /= (q1.w + 1e-16f);
  po[0] = p0;
  po[1] = p1;
}

// ---------------------------------------------------------------------------
extern "C" void kernel_launch(void* const* d_in, const int* in_sizes, int n_in,
                              void* d_out, int out_size, void* d_ws, size_t ws_size,
                              hipStream_t stream) {
  const float* x    = (const float*)d_in[0];   // (B,N,256) f32
  const int*   edge = (const int*)d_in[1];     // (B,2,E)  i32
  const float* W    = (const float*)d_in[2];   // (256,256) f32
  const float* a    = (const float*)d_in[3];   // (64,)    f32

  // Output tuple: attention (B,E,H) then wx (B,N,256), concatenated flat.
  float* att = (float*)d_out;
  float* wx  = (float*)d_out + (size_t)B_ * E_ * H_;

  // Scratch: s_src | s_dst | segmax(keys) | segsum | Wt  — ~5.4 MB
  float*    s_src  = (float*)d_ws;
  float*    s_dst  = s_src + (size_t)B_ * N_ * H_;
  unsigned* segmax = (unsigned*)(s_dst + (size_t)B_ * N_ * H_);
  float*    segsum = (float*)(segmax + (size_t)B_ * N_ * H_);
  float*    Wt     = segsum + (size_t)B_ * N_ * H_;

  const int nNodeH = B_ * N_ * H_;                 // 320000
  const int nEdges = B_ * E_;                      // 1280000

  gat_transpose_w<<<(ATTD_ * INF_ + 255) / 256, 256, 0, stream>>>(W, Wt);
  gat_gemm_wx<<<(B_ * N_) / 32, 256, 0, stream>>>(x, Wt, wx);
  gat_init_seg<<<(nNodeH + 255) / 256, 256, 0, stream>>>(segmax, segsum);
  gat_node_scores<<<(nNodeH + 255) / 256, 256, 0, stream>>>(wx, a, s_src, s_dst);
  gat_edge_score<<<(nEdges + 255) / 256, 256, 0, stream>>>(edge, s_src, s_dst, att, segmax);
  gat_edge_exp<<<(nEdges + 255) / 256, 256, 0, stream>>>(edge, segmax, att, segsum);
  gat_edge_norm<<<(nEdges + 255) / 256, 256, 0, stream>>>(edge, segsum, att);
}